// OrthoSheafs_2594160246966
// MI455X (gfx1250) — compile-verified
//
#include <hip/hip_runtime.h>
#include <math.h>

// Problem constants (match reference)
#define Nn   20000
#define Ee   5000
#define NNZv 100000
#define Ff   64
#define Hh   64
#define Dd   4
#define Cc   40

typedef float v2f __attribute__((ext_vector_type(2)));
typedef float v8f __attribute__((ext_vector_type(8)));

// ---------------------------------------------------------------------------
// Generic fp32 WMMA GEMM: C[M x Nc] = A[M x K] @ B[K x Nc], row-major.
// One wave per 16x16 output tile; K stepped by 4 via V_WMMA_F32_16X16X4_F32.
// Branchless edge handling: addresses are clamped in-range, loads issued
// unconditionally, values zeroed by v_cndmask selects (EXEC stays all-ones,
// which WMMA requires, and no saveexec branches around loads).
// A layout (16x4 f32): lane&15 = row M; lane-half selects K pair {0,1}/{2,3}.
// B layout (4x16 f32): lane&15 = col N; lane-half selects K rows {0,1}/{2,3}.
// C/D layout: vgpr j -> row j + 8*half, col = lane&15.
// ---------------------------------------------------------------------------
template <int K>
__global__ __launch_bounds__(32)
void gemm_f32_wmma_kernel(const float* __restrict__ A, const float* __restrict__ B,
                          float* __restrict__ C, int M, int Nc) {
    const int tm   = blockIdx.x;
    const int tn   = blockIdx.y;
    const int lane = threadIdx.x;
    const int half = lane >> 4;        // 0 or 1
    const int l15  = lane & 15;
    const int m = tm * 16 + l15;       // A row served by this lane
    const int n = tn * 16 + l15;       // B col served by this lane

    const bool mv = (m < M);
    const bool nv = (n < Nc);
    const int  mc = mv ? m : (M - 1);  // clamped (always valid) addresses
    const int  nc = nv ? n : (Nc - 1);

    const float* Ap = A + (size_t)mc * K + 2 * half;   // two contiguous K elems
    const float* Bp = B + (size_t)(2 * half) * Nc + nc;

    v8f acc = {};
#pragma unroll 8
    for (int k = 0; k < K; k += 4) {
        v2f a = *(const v2f*)(Ap + k);           // global_load_b64
        v2f b;
        b.x = Bp[(size_t)k * Nc];
        b.y = Bp[(size_t)(k + 1) * Nc];
        a.x = mv ? a.x : 0.0f;                   // v_cndmask, no branch
        a.y = mv ? a.y : 0.0f;
        b.x = nv ? b.x : 0.0f;
        b.y = nv ? b.y : 0.0f;
        acc = __builtin_amdgcn_wmma_f32_16x16x4_f32(
            /*neg_a=*/false, a, /*neg_b=*/false, b,
            /*c_mod=*/(short)0, acc, /*reuse_a=*/false, /*reuse_b=*/false);
    }
#pragma unroll
    for (int j = 0; j < 8; ++j) {
        const int r = tm * 16 + j + 8 * half;
        if (r < M && n < Nc) C[(size_t)r * Nc + n] = acc[j];
    }
}

// ---------------------------------------------------------------------------
// Degrees: deg_v[row[k]] += 1, deg_e[col[k]] += 1
// ---------------------------------------------------------------------------
__global__ void degrees_kernel(const int* __restrict__ row, const int* __restrict__ col,
                               float* __restrict__ deg_v, float* __restrict__ deg_e, int nnz) {
    int k = blockIdx.x * blockDim.x + threadIdx.x;
    if (k >= nnz) return;
    atomicAdd(&deg_v[row[k]], 1.0f);
    atomicAdd(&deg_e[col[k]], 1.0f);
}

// he[col[k], :] += x[row[k], :]   (thread per (k, f/4): b128 gather, 4 atomics)
__global__ void he_scatter_kernel(const float* __restrict__ x,
                                  const int* __restrict__ row, const int* __restrict__ col,
                                  float* __restrict__ he, int nnz) {
    int idx = blockIdx.x * blockDim.x + threadIdx.x;
    if (idx >= nnz * (Ff / 4)) return;
    int k = idx >> 4, f4 = (idx & 15) << 2;
    const float4 v = *(const float4*)(x + (size_t)row[k] * Ff + f4);
    float* dst = he + (size_t)col[k] * Ff + f4;
    atomicAdd(dst + 0, v.x);
    atomicAdd(dst + 1, v.y);
    atomicAdd(dst + 2, v.z);
    atomicAdd(dst + 3, v.w);
}

// he[e,:] /= max(deg_e[e], 1)   (float4 per thread)
__global__ void he_norm_kernel(float* __restrict__ he, const float* __restrict__ deg_e) {
    int idx = blockIdx.x * blockDim.x + threadIdx.x;
    if (idx >= Ee * (Ff / 4)) return;
    float r = 1.0f / fmaxf(deg_e[idx >> 4], 1.0f);
    float4* p = (float4*)he + idx;
    float4 v = *p;
    v.x *= r; v.y *= r; v.z *= r; v.w *= r;
    *p = v;
}

// dst[n, f] = mean over d of src[n, d, f]  (src M x 256, dst M x 64; float4)
__global__ void mean_d_kernel(const float* __restrict__ src, float* __restrict__ dst, int M) {
    int idx = blockIdx.x * blockDim.x + threadIdx.x;
    if (idx >= M * (Ff / 4)) return;
    int n = idx >> 4, f4 = (idx & 15) << 2;
    const float* s = src + (size_t)n * (Dd * Ff) + f4;
    const float4 a = *(const float4*)(s);
    const float4 b = *(const float4*)(s + 64);
    const float4 c = *(const float4*)(s + 128);
    const float4 d = *(const float4*)(s + 192);
    float4 o;
    o.x = 0.25f * (a.x + b.x + c.x + d.x);
    o.y = 0.25f * (a.y + b.y + c.y + d.y);
    o.z = 0.25f * (a.z + b.z + c.z + d.z);
    o.w = 0.25f * (a.w + b.w + c.w + d.w);
    *((float4*)dst + idx) = o;
}

// ---------------------------------------------------------------------------
// hs = [xmean[row], emean[col]] @ W_sheaf ; p = tanh(hs) ;
// Q = householder(p, 4)  -> Qb[k, 4, 4]   (thread per nnz, W_sheaf in LDS)
// ---------------------------------------------------------------------------
__global__ void sheaf_q_kernel(const float* __restrict__ xmean, const float* __restrict__ emean,
                               const int* __restrict__ row, const int* __restrict__ col,
                               const float* __restrict__ Wsheaf, float* __restrict__ Qb, int nnz) {
    __shared__ float Ws[2 * Ff * 6];
    for (int i = threadIdx.x; i < 2 * Ff * 6; i += blockDim.x) Ws[i] = Wsheaf[i];
    __syncthreads();

    int k = blockIdx.x * blockDim.x + threadIdx.x;
    if (k >= nnz) return;
    const float* xm = xmean + (size_t)row[k] * Ff;
    const float* em = emean + (size_t)col[k] * Ff;

    float hs[6] = {0, 0, 0, 0, 0, 0};
    for (int f = 0; f < Ff; ++f) {
        float a = xm[f];
#pragma unroll
        for (int j = 0; j < 6; ++j) hs[j] += a * Ws[f * 6 + j];
    }
    for (int f = 0; f < Ff; ++f) {
        float b = em[f];
#pragma unroll
        for (int j = 0; j < 6; ++j) hs[j] += b * Ws[(Ff + f) * 6 + j];
    }

    float p[6];
#pragma unroll
    for (int j = 0; j < 6; ++j) p[j] = tanhf(hs[j]);

    // A = strictly-lower-tri fill at tril_indices(4,-1) order:
    // (1,0),(2,0),(2,1),(3,0),(3,1),(3,2)
    float Am[4][4] = {};
    Am[1][0] = p[0]; Am[2][0] = p[1]; Am[2][1] = p[2];
    Am[3][0] = p[3]; Am[3][1] = p[4]; Am[3][2] = p[5];

    float tau[4];
#pragma unroll
    for (int j = 0; j < 4; ++j) {
        float s = 0.0f;
#pragma unroll
        for (int i = 0; i < 4; ++i) s += Am[i][j] * Am[i][j];
        tau[j] = 2.0f / (1.0f + s);
    }

    float V[4][4], Q[4][4];
#pragma unroll
    for (int i = 0; i < 4; ++i)
#pragma unroll
        for (int j = 0; j < 4; ++j) {
            V[i][j] = Am[i][j] + (i == j ? 1.0f : 0.0f);
            Q[i][j] = (i == j) ? 1.0f : 0.0f;
        }

#pragma unroll
    for (int j = 0; j < 4; ++j) {
        float v[4], Qv[4];
#pragma unroll
        for (int i = 0; i < 4; ++i) v[i] = V[i][j];
#pragma unroll
        for (int i = 0; i < 4; ++i) {
            float s = 0.0f;
#pragma unroll
            for (int q = 0; q < 4; ++q) s += Q[i][q] * v[q];
            Qv[i] = s;
        }
#pragma unroll
        for (int i = 0; i < 4; ++i)
#pragma unroll
            for (int q = 0; q < 4; ++q) Q[i][q] -= tau[j] * Qv[i] * v[q];
    }

    float* qo = Qb + (size_t)k * 16;
#pragma unroll
    for (int i = 0; i < 4; ++i)
#pragma unroll
        for (int j = 0; j < 4; ++j) qo[i * 4 + j] = Q[i][j];
}

// ---------------------------------------------------------------------------
// sheaf_conv pass A: m[k,q,f] = sum_p Q[k,p,q] * t0[row[k],p,f];
//                    ebuf[col[k],q,f] += m   (thread per (k, f/4))
// ---------------------------------------------------------------------------
__global__ void conv_scatter_kernel(const float* __restrict__ t0, const float* __restrict__ Qb,
                                    const int* __restrict__ row, const int* __restrict__ col,
                                    float* __restrict__ ebuf, int nnz) {
    int idx = blockIdx.x * blockDim.x + threadIdx.x;
    if (idx >= nnz * (Hh / 4)) return;
    int k = idx >> 4, f4 = (idx & 15) << 2;
    int r = row[k], c = col[k];
    const float* Q = Qb + (size_t)k * 16;
    float4 xp[4];
#pragma unroll
    for (int p = 0; p < 4; ++p)
        xp[p] = *(const float4*)(t0 + (size_t)r * (Dd * Hh) + p * Hh + f4);
#pragma unroll
    for (int q = 0; q < 4; ++q) {
        float4 m = {0, 0, 0, 0};
#pragma unroll
        for (int p = 0; p < 4; ++p) {
            float w = Q[p * 4 + q];
            m.x += w * xp[p].x; m.y += w * xp[p].y;
            m.z += w * xp[p].z; m.w += w * xp[p].w;
        }
        float* dst = ebuf + (size_t)c * (Dd * Hh) + q * Hh + f4;
        atomicAdd(dst + 0, m.x);
        atomicAdd(dst + 1, m.y);
        atomicAdd(dst + 2, m.z);
        atomicAdd(dst + 3, m.w);
    }
}

// ebuf[e,:,:] *= Be[e],  Be = deg_e>0 ? 1/(D*deg_e) : 0   (float4)
__global__ void escale_kernel(float* __restrict__ ebuf, const float* __restrict__ deg_e) {
    int idx = blockIdx.x * blockDim.x + threadIdx.x;
    if (idx >= Ee * (Dd * Hh / 4)) return;
    float d = deg_e[idx >> 6];
    float s = (d > 0.0f) ? 1.0f / (Dd * d) : 0.0f;
    float4* p = (float4*)ebuf + idx;
    float4 v = *p;
    v.x *= s; v.y *= s; v.z *= s; v.w *= s;
    *p = v;
}

// sheaf_conv pass B: m2[k,p,f] = sum_q Q[k,p,q] * ebuf[col[k],q,f];
//                    nodebuf[row[k],p,f] += m2   (thread per (k, f/4))
__global__ void conv_gather_kernel(const float* __restrict__ ebuf, const float* __restrict__ Qb,
                                   const int* __restrict__ row, const int* __restrict__ col,
                                   float* __restrict__ nodebuf, int nnz) {
    int idx = blockIdx.x * blockDim.x + threadIdx.x;
    if (idx >= nnz * (Hh / 4)) return;
    int k = idx >> 4, f4 = (idx & 15) << 2;
    int r = row[k], c = col[k];
    const float* Q = Qb + (size_t)k * 16;
    float4 eq[4];
#pragma unroll
    for (int q = 0; q < 4; ++q)
        eq[q] = *(const float4*)(ebuf + (size_t)c * (Dd * Hh) + q * Hh + f4);
#pragma unroll
    for (int p = 0; p < 4; ++p) {
        float4 m = {0, 0, 0, 0};
#pragma unroll
        for (int q = 0; q < 4; ++q) {
            float w = Q[p * 4 + q];
            m.x += w * eq[q].x; m.y += w * eq[q].y;
            m.z += w * eq[q].z; m.w += w * eq[q].w;
        }
        float* dst = nodebuf + (size_t)r * (Dd * Hh) + p * Hh + f4;
        atomicAdd(dst + 0, m.x);
        atomicAdd(dst + 1, m.y);
        atomicAdd(dst + 2, m.z);
        atomicAdd(dst + 3, m.w);
    }
}

// nodebuf *= Dv  (+ optional ELU), in place   (float4)
__global__ void node_finish_kernel(float* __restrict__ buf, const float* __restrict__ deg_v,
                                   int apply_elu) {
    int idx = blockIdx.x * blockDim.x + threadIdx.x;
    if (idx >= Nn * (Dd * Hh / 4)) return;
    float d = deg_v[idx >> 6];
    float s = (d > 0.0f) ? 1.0f / (Dd * d) : 0.0f;
    float4* p = (float4*)buf + idx;
    float4 v = *p;
    v.x *= s; v.y *= s; v.z *= s; v.w *= s;
    if (apply_elu) {
        v.x = (v.x > 0.0f) ? v.x : expm1f(v.x);
        v.y = (v.y > 0.0f) ? v.y : expm1f(v.y);
        v.z = (v.z > 0.0f) ? v.z : expm1f(v.z);
        v.w = (v.w > 0.0f) ? v.w : expm1f(v.w);
    }
    *p = v;
}

// ---------------------------------------------------------------------------
extern "C" void kernel_launch(void* const* d_in, const int* in_sizes, int n_in,
                              void* d_out, int out_size, void* d_ws, size_t ws_size,
                              hipStream_t stream) {
    const float* x       = (const float*)d_in[0];
    const int*   ei      = (const int*)d_in[1];
    const float* W_lin   = (const float*)d_in[2];
    const float* W_sheaf = (const float*)d_in[3];
    const float* W_conv0 = (const float*)d_in[4];
    const float* W_conv1 = (const float*)d_in[5];
    const float* W_lin2  = (const float*)d_in[6];
    float* out = (float*)d_out;

    const int* row = ei;            // edge_index[0]
    const int* col = ei + NNZv;     // edge_index[1]

    // Workspace carve (floats). h2 aliases xb (xb dead before conv1 output).
    float* p = (float*)d_ws;
    float* deg_v = p; p += Nn;
    float* deg_e = p; p += Ee;
    float* he    = p; p += (size_t)Ee * Ff;
    float* xb    = p; p += (size_t)Nn * Dd * Ff;   // later reused as h2
    float* eb    = p; p += (size_t)Ee * Dd * Ff;
    float* xmean = p; p += (size_t)Nn * Ff;
    float* emean = p; p += (size_t)Ee * Ff;
    float* Qb    = p; p += (size_t)NNZv * 16;
    float* t0    = p; p += (size_t)Nn * Dd * Hh;
    float* ebuf  = p; p += (size_t)Ee * Dd * Hh;
    float* h1    = p; p += (size_t)Nn * Dd * Hh;
    float* h2    = xb;

    const int B = 256;
    auto cdiv = [](int a, int b) { return (a + b - 1) / b; };

    // --- degrees + hyperedge features -------------------------------------
    hipMemsetAsync(deg_v, 0, (size_t)(Nn + Ee) * sizeof(float), stream);  // deg_v+deg_e contiguous
    hipMemsetAsync(he, 0, (size_t)Ee * Ff * sizeof(float), stream);
    degrees_kernel<<<cdiv(NNZv, B), B, 0, stream>>>(row, col, deg_v, deg_e, NNZv);
    he_scatter_kernel<<<cdiv(NNZv * (Ff / 4), B), B, 0, stream>>>(x, row, col, he, NNZv);
    he_norm_kernel<<<cdiv(Ee * (Ff / 4), B), B, 0, stream>>>(he, deg_e);

    // --- stalk projections: xb = x@W_lin, eb = he@W_lin (fp32 WMMA) -------
    {
        dim3 g(cdiv(Nn, 16), cdiv(Dd * Ff, 16));
        gemm_f32_wmma_kernel<Ff><<<g, 32, 0, stream>>>(x, W_lin, xb, Nn, Dd * Ff);
    }
    {
        dim3 g(cdiv(Ee, 16), cdiv(Dd * Ff, 16));
        gemm_f32_wmma_kernel<Ff><<<g, 32, 0, stream>>>(he, W_lin, eb, Ee, Dd * Ff);
    }

    // --- sheaf maps: means -> hs -> tanh -> Householder Q -----------------
    mean_d_kernel<<<cdiv(Nn * (Ff / 4), B), B, 0, stream>>>(xb, xmean, Nn);
    mean_d_kernel<<<cdiv(Ee * (Ff / 4), B), B, 0, stream>>>(eb, emean, Ee);
    sheaf_q_kernel<<<cdiv(NNZv, B), B, 0, stream>>>(xmean, emean, row, col, W_sheaf, Qb, NNZv);

    // --- conv0: h1 = elu(sheaf_conv(xb, W_conv0)) -------------------------
    {
        dim3 g(cdiv(Nn * Dd, 16), cdiv(Hh, 16));   // (N*D x 64) @ (64 x 64)
        gemm_f32_wmma_kernel<Ff><<<g, 32, 0, stream>>>(xb, W_conv0, t0, Nn * Dd, Hh);
    }
    hipMemsetAsync(ebuf, 0, (size_t)Ee * Dd * Hh * sizeof(float), stream);
    conv_scatter_kernel<<<cdiv(NNZv * (Hh / 4), B), B, 0, stream>>>(t0, Qb, row, col, ebuf, NNZv);
    escale_kernel<<<cdiv(Ee * (Dd * Hh / 4), B), B, 0, stream>>>(ebuf, deg_e);
    hipMemsetAsync(h1, 0, (size_t)Nn * Dd * Hh * sizeof(float), stream);
    conv_gather_kernel<<<cdiv(NNZv * (Hh / 4), B), B, 0, stream>>>(ebuf, Qb, row, col, h1, NNZv);
    node_finish_kernel<<<cdiv(Nn * (Dd * Hh / 4), B), B, 0, stream>>>(h1, deg_v, /*elu=*/1);

    // --- conv1: h2 = sheaf_conv(h1, W_conv1) ------------------------------
    {
        dim3 g(cdiv(Nn * Dd, 16), cdiv(Hh, 16));
        gemm_f32_wmma_kernel<Hh><<<g, 32, 0, stream>>>(h1, W_conv1, t0, Nn * Dd, Hh);
    }
    hipMemsetAsync(ebuf, 0, (size_t)Ee * Dd * Hh * sizeof(float), stream);
    conv_scatter_kernel<<<cdiv(NNZv * (Hh / 4), B), B, 0, stream>>>(t0, Qb, row, col, ebuf, NNZv);
    escale_kernel<<<cdiv(Ee * (Dd * Hh / 4), B), B, 0, stream>>>(ebuf, deg_e);
    hipMemsetAsync(h2, 0, (size_t)Nn * Dd * Hh * sizeof(float), stream);
    conv_gather_kernel<<<cdiv(NNZv * (Hh / 4), B), B, 0, stream>>>(ebuf, Qb, row, col, h2, NNZv);
    node_finish_kernel<<<cdiv(Nn * (Dd * Hh / 4), B), B, 0, stream>>>(h2, deg_v, /*elu=*/0);

    // --- classifier: out = h2.reshape(N,256) @ W_lin2 (256 x 40) ----------
    {
        dim3 g(cdiv(Nn, 16), cdiv(Cc, 16));
        gemm_f32_wmma_kernel<Dd * Hh><<<g, 32, 0, stream>>>(h2, W_lin2, out, Nn, Cc);
    }
}